// MultiConvPerm_16466904613016
// MI455X (gfx1250) — compile-verified
//
#include <hip/hip_runtime.h>
#include <math.h>

typedef __attribute__((ext_vector_type(2))) float v2f;
typedef __attribute__((ext_vector_type(8))) float v8f;

#define NWT   12
#define NTAP  72
#define SLEN  57

// kernel size per weight bank
constexpr int KW[NWT] = {3,3,3,5,5,5,7,7,7,9,9,9};

// Flattened tap tables: for each of the 72 taps, which weight bank (I),
// which tap index inside that bank (J), and which position s in [0,57) the
// dilated+center-padded tap lands on (S).  S = p_i + d_i*j, p_i = (57-L_i)/2.
constexpr int TAP_I[NTAP] = {
  0,0,0,  1,1,1,  2,2,2,
  3,3,3,3,3,  4,4,4,4,4,  5,5,5,5,5,
  6,6,6,6,6,6,6,  7,7,7,7,7,7,7,  8,8,8,8,8,8,8,
  9,9,9,9,9,9,9,9,9,  10,10,10,10,10,10,10,10,10,  11,11,11,11,11,11,11,11,11
};
constexpr int TAP_J[NTAP] = {
  0,1,2,  0,1,2,  0,1,2,
  0,1,2,3,4,  0,1,2,3,4,  0,1,2,3,4,
  0,1,2,3,4,5,6,  0,1,2,3,4,5,6,  0,1,2,3,4,5,6,
  0,1,2,3,4,5,6,7,8,  0,1,2,3,4,5,6,7,8,  0,1,2,3,4,5,6,7,8
};
constexpr int TAP_S[NTAP] = {
  27,28,29,  25,28,31,  21,28,35,
  26,27,28,29,30,  22,25,28,31,34,  14,21,28,35,42,
  25,26,27,28,29,30,31,  19,22,25,28,31,34,37,  7,14,21,28,35,42,49,
  24,25,26,27,28,29,30,31,32,  16,19,22,25,28,31,34,37,40,  0,7,14,21,28,35,42,49,56
};

// Template-recursive K-loop so every tap-table lookup is a compile-time
// literal (no constant-memory gathers); only the lane-half select survives.
template <int KK>
__device__ __forceinline__ void mix_steps(const float* const (&wp)[NWT],
                                          const float (&coef)[NWT],
                                          long long pairA, bool hiH, int n,
                                          v8f (&acc)[4])
{
  if constexpr (KK < 18) {
    constexpr int k0  = KK * 4;
    constexpr int iX0 = TAP_I[k0 + 0], iX1 = TAP_I[k0 + 2];
    constexpr int iY0 = TAP_I[k0 + 1], iY1 = TAP_I[k0 + 3];
    constexpr int jX0 = TAP_J[k0 + 0], jX1 = TAP_J[k0 + 2];
    constexpr int jY0 = TAP_J[k0 + 1], jY1 = TAP_J[k0 + 3];
    constexpr int sX0 = TAP_S[k0 + 0], sX1 = TAP_S[k0 + 2];
    constexpr int sY0 = TAP_S[k0 + 1], sY1 = TAP_S[k0 + 3];
    constexpr int kX0 = KW[iX0], kX1 = KW[iX1];
    constexpr int kY0 = KW[iY0], kY1 = KW[iY1];

    const float*    pX = hiH ? wp[iX1] : wp[iX0];
    const float*    pY = hiH ? wp[iY1] : wp[iY0];
    const long long oX = hiH ? (pairA * kX1 + jX1) : (pairA * kX0 + jX0);
    const long long oY = hiH ? (pairA * kY1 + jY1) : (pairA * kY0 + jY0);
    const float     cX = hiH ? coef[iX1] : coef[iX0];
    const float     cY = hiH ? coef[iY1] : coef[iY0];
    const int       sX = hiH ? sX1 : sX0;
    const int       sY = hiH ? sY1 : sY0;

    v2f a;
    a.x = pX[oX];   // A[m][k0+hk]
    a.y = pY[oY];   // A[m][k0+hk+1]

    #pragma unroll
    for (int tt = 0; tt < 4; ++tt) {
      v2f b;
      b.x = (sX == tt * 16 + n) ? cX : 0.f;   // B[k0+hk][tt*16+n]
      b.y = (sY == tt * 16 + n) ? cY : 0.f;
      acc[tt] = __builtin_amdgcn_wmma_f32_16x16x4_f32(
          false, a, false, b, (short)0, acc[tt], false, false);
    }
    mix_steps<KK + 1>(wp, coef, pairA, hiH, n, acc);
  }
}

__global__ __launch_bounds__(256)
void multiconv_wmma_kernel(const float* __restrict__ w0,  const float* __restrict__ w1,
                           const float* __restrict__ w2,  const float* __restrict__ w3,
                           const float* __restrict__ w4,  const float* __restrict__ w5,
                           const float* __restrict__ w6,  const float* __restrict__ w7,
                           const float* __restrict__ w8,  const float* __restrict__ w9,
                           const float* __restrict__ w10, const float* __restrict__ w11,
                           const float* __restrict__ alpha, const float* __restrict__ gumbels,
                           float* __restrict__ out, long long pairCount, int isz)
{
  const float* wp[NWT] = {w0,w1,w2,w3,w4,w5,w6,w7,w8,w9,w10,w11};

  const int  tid  = threadIdx.x;
  const int  lane = tid & 31;
  const int  wave = tid >> 5;
  const bool hiH  = (lane >= 16);    // which 16-lane half of the wave32
  const int  n    = lane & 15;       // N (column) index for B/C/D, M row for A

  // ---- gumbel-softmax coefficients (wave-uniform; scalarizes) ----
  float av[NWT], gv[NWT];
  #pragma unroll
  for (int i = 0; i < NWT; ++i) { av[i] = alpha[i]; gv[i] = gumbels[i]; }
  float m1 = av[0];
  #pragma unroll
  for (int i = 1; i < NWT; ++i) m1 = fmaxf(m1, av[i]);
  float s1 = 0.f;
  #pragma unroll
  for (int i = 0; i < NWT; ++i) s1 += expf(av[i] - m1);
  const float lse = m1 + logf(s1);
  float xv[NWT];
  #pragma unroll
  for (int i = 0; i < NWT; ++i) xv[i] = av[i] - lse + gv[i];
  float m2 = xv[0];
  #pragma unroll
  for (int i = 1; i < NWT; ++i) m2 = fmaxf(m2, xv[i]);
  float ev[NWT]; float s2 = 0.f;
  #pragma unroll
  for (int i = 0; i < NWT; ++i) { ev[i] = expf(xv[i] - m2); s2 += ev[i]; }
  const float invs = 1.f / s2;
  float coef[NWT];
  #pragma unroll
  for (int i = 0; i < NWT; ++i) coef[i] = ev[i] * invs;

  // ---- WMMA mixture GEMM: (16 pairs x 72 taps) x (72 taps x 64 s) ----
  const long long pairBase = ((long long)blockIdx.x * 8 + wave) * 16;
  long long pairA = pairBase + n;
  if (pairA >= pairCount) pairA = pairCount - 1;   // clamp gathers (results unused)

  v8f acc[4] = {};   // 4 N-tiles of 16 -> s in [0,64)
  mix_steps<0>(wp, coef, pairA, hiH, n, acc);

  // ---- stage D tiles to LDS: stage[wave][m*64 + s] ----
  __shared__ float stage[8][16 * 64];
  #pragma unroll
  for (int tt = 0; tt < 4; ++tt) {
    #pragma unroll
    for (int r = 0; r < 8; ++r) {
      const int mm = r + (hiH ? 8 : 0);
      stage[wave][mm * 64 + tt * 16 + n] = acc[tt][r];
    }
  }
  __syncthreads();

  // ---- stream output: out[pair][t] = (u<57) ? acc[pair][u] : 0, u=(28-t) mod isz ----
  // For isz >= 256 only the first and last 128-element chunks of each row can
  // be nonzero (t <= 28 or t >= isz-28); everything in between is a pure
  // zero-fill with b128 stores.
  const float* st = stage[wave];
  if ((isz & 127) == 0 && isz >= 256) {
    const int t4 = lane * 4;
    const float4 z4 = make_float4(0.f, 0.f, 0.f, 0.f);
    for (int mm = 0; mm < 16; ++mm) {
      const long long pr = pairBase + mm;
      if (pr >= pairCount) break;
      float* op = out + pr * (long long)isz;
      const float* row = st + mm * 64;
      { // head chunk: t in [0,128), u = 28 - t
        float4 v;
        v.x = (t4     <= 28) ? row[28 - t4] : 0.f;
        v.y = (t4 + 1 <= 28) ? row[27 - t4] : 0.f;
        v.z = (t4 + 2 <= 28) ? row[26 - t4] : 0.f;
        v.w = (t4 + 3 <= 28) ? row[25 - t4] : 0.f;
        *(float4*)(op + t4) = v;
      }
      for (int tb = 128; tb < isz - 128; tb += 128)   // middle: zeros
        *(float4*)(op + tb + t4) = z4;
      { // tail chunk: t = isz-128+t4+e, u = 156 - t4 - e (always >= 29)
        const int u0 = 156 - t4;
        float4 v;
        v.x = (u0     < SLEN) ? row[u0]     : 0.f;
        v.y = (u0 - 1 < SLEN) ? row[u0 - 1] : 0.f;
        v.z = (u0 - 2 < SLEN) ? row[u0 - 2] : 0.f;
        v.w = (u0 - 3 < SLEN) ? row[u0 - 3] : 0.f;
        *(float4*)(op + (isz - 128) + t4) = v;
      }
    }
  } else {
    for (int mm = 0; mm < 16; ++mm) {
      const long long pr = pairBase + mm;
      if (pr >= pairCount) break;
      float* op = out + pr * (long long)isz;
      for (int t = lane; t < isz; t += 32) {
        int u = (28 - t) % isz; if (u < 0) u += isz;
        op[t] = (u < SLEN) ? st[mm * 64 + u] : 0.f;
      }
    }
  }
}

extern "C" void kernel_launch(void* const* d_in, const int* in_sizes, int n_in,
                              void* d_out, int out_size, void* d_ws, size_t ws_size,
                              hipStream_t stream) {
  (void)n_in; (void)d_ws; (void)ws_size;
  const float* w[NWT];
  for (int i = 0; i < NWT; ++i) w[i] = (const float*)d_in[i];
  const float* alpha   = (const float*)d_in[12];
  const float* gumbels = (const float*)d_in[13];

  const long long pairs = in_sizes[0] / KW[0];          // CO*CI = 65536
  const int isz = (int)((long long)out_size / pairs);   // 1024
  const int blocks = (int)((pairs + 127) / 128);        // 128 pairs per 256-thread block

  multiconv_wmma_kernel<<<blocks, 256, 0, stream>>>(
      w[0], w[1], w[2], w[3], w[4], w[5], w[6], w[7], w[8], w[9], w[10], w[11],
      alpha, gumbels, (float*)d_out, pairs, isz);
}